// PlayCell_30116310680361
// MI455X (gfx1250) — compile-verified
//
#include <hip/hip_runtime.h>
#include <stdint.h>

// Hysteresis play-operator recurrence  out[t] = clamp(out[t-1], y[t], y[t]+W)
// parallelized as an associative clamp-composition scan over (lo,hi) pairs.
// out[t] = clamp(state, Lo[t], Hi[t]).
//
// CDNA5 paths used:
//   * global_load_async_to_lds_b128 + s_wait_asynccnt  (ASYNCcnt DMA staging)
//   * wave32 shuffle scans (warpSize==32 on gfx1250)
//   * non-temporal stores so the 64MB output stream doesn't evict x from the
//     192MB L2 between pass 1 and pass 3.

#define PLAY_WIDTH 1.0f
#define BLOCK 256              // 8 wave32 waves
#define NWAVES (BLOCK / 32)
#define ITEMS 16               // contiguous elements per thread (scan order!)
#define CHUNK (BLOCK * ITEMS)  // 4096 elements = 16 KB per block

struct Seg { float lo, hi; };  // x -> min(max(x, lo), hi)

__device__ __forceinline__ Seg seg_ident() {
  return Seg{-__builtin_inff(), __builtin_inff()};
}

// apply a first, then b (associative, NOT commutative)
__device__ __forceinline__ Seg seg_combine(Seg a, Seg b) {
  Seg r;
  r.lo = fminf(fmaxf(a.lo, b.lo), b.hi);
  r.hi = fminf(fmaxf(a.hi, b.lo), b.hi);
  return r;
}

// ---- CDNA5 async global->LDS copy (ASYNCcnt path) -------------------------
typedef __attribute__((address_space(3))) char lds_char_t;

__device__ __forceinline__ uint32_t lds_byte_offset(void* p) {
  return (uint32_t)(uintptr_t)(lds_char_t*)p;   // AS(3) ptr == raw LDS offset
}

__device__ __forceinline__ void async_copy_b128(void* lds_dst, const void* gsrc) {
  uint32_t loff = lds_byte_offset(lds_dst);
  asm volatile("global_load_async_to_lds_b128 %0, %1, off"
               :: "v"(loff), "v"(gsrc)
               : "memory");
}

__device__ __forceinline__ void wait_async() {
  asm volatile("s_wait_asynccnt 0" ::: "memory");
}

// Stage one CHUNK of x into LDS (per-lane 16B async DMA, 4 block-wide rounds).
__device__ __forceinline__ void stage_chunk(float* s_x, const float* __restrict__ x,
                                            long long base, long long T) {
  const int tid = threadIdx.x;
  if (base + CHUNK <= T) {
#pragma unroll
    for (int i = 0; i < (CHUNK * 4) / (BLOCK * 16); ++i) {  // = 4
      const int boff = (i * BLOCK + tid) * 16;
      async_copy_b128((char*)s_x + boff, (const char*)(x + base) + boff);
    }
    wait_async();          // each wave drains its own ASYNCcnt ...
  } else {
    for (int i = tid; i < CHUNK; i += BLOCK) {
      long long idx = base + i;
      s_x[i] = (idx < T) ? x[idx] : 0.0f;   // guarded again at compose time
    }
  }
  __syncthreads();         // ... then barrier publishes all waves' data
}

// ---- ordered wave32 inclusive shuffle scan --------------------------------
__device__ __forceinline__ Seg wave_incl_scan(Seg a, int lane) {
#pragma unroll
  for (int s = 1; s < 32; s <<= 1) {
    float plo = __shfl_up(a.lo, s, 32);
    float phi = __shfl_up(a.hi, s, 32);
    if (lane >= s) a = seg_combine(Seg{plo, phi}, a);
  }
  return a;
}

// Block scan over per-thread aggregates: returns this thread's inclusive
// composition; *excl gets the exclusive one. One barrier, 8-entry LDS.
__device__ __forceinline__ Seg block_scan(Seg a, float2* s_wave, Seg* excl) {
  const int tid  = threadIdx.x;
  const int lane = tid & 31;
  const int wid  = tid >> 5;

  Seg incl = wave_incl_scan(a, lane);
  if (lane == 31) s_wave[wid] = make_float2(incl.lo, incl.hi);
  __syncthreads();

  Seg wp = seg_ident();                       // prefix of earlier waves
#pragma unroll
  for (int w = 0; w < NWAVES; ++w) {
    if (w < wid) {
      float2 t = s_wave[w];
      wp = seg_combine(wp, Seg{t.x, t.y});
    }
  }
  float elo = __shfl_up(incl.lo, 1, 32);
  float ehi = __shfl_up(incl.hi, 1, 32);
  *excl = (lane == 0) ? wp : seg_combine(wp, Seg{elo, ehi});
  return seg_combine(wp, incl);
}

// ---- pass 1: per-block aggregates -----------------------------------------
__global__ void __launch_bounds__(BLOCK)
k_block_agg(const float* __restrict__ x, const float* __restrict__ kw,
            float2* __restrict__ aggs, long long T) {
  __shared__ float  s_x[CHUNK];
  __shared__ float2 s_wave[NWAVES];
  const int tid = threadIdx.x;
  const long long base = (long long)blockIdx.x * CHUNK;

  stage_chunk(s_x, x, base, T);
  const float k = kw[0];

  Seg a = seg_ident();
#pragma unroll
  for (int j = 0; j < ITEMS; ++j) {
    long long idx = base + (long long)tid * ITEMS + j;
    if (idx < T) {
      float y = k * s_x[tid * ITEMS + j];
      a = seg_combine(a, Seg{y, y + PLAY_WIDTH});
    }
  }
  Seg excl;
  Seg incl = block_scan(a, s_wave, &excl);
  if (tid == BLOCK - 1) aggs[blockIdx.x] = make_float2(incl.lo, incl.hi);
}

// ---- pass 2: scan block aggregates (one block) ----------------------------
__global__ void __launch_bounds__(BLOCK)
k_scan_aggs(const float2* __restrict__ aggs, float2* __restrict__ pref, int nb) {
  __shared__ float2 s_wave[NWAVES];
  const int tid = threadIdx.x;
  const int G = (nb + BLOCK - 1) / BLOCK;     // contiguous run per thread

  Seg a = seg_ident();
  for (int g = 0; g < G; ++g) {
    int i = tid * G + g;
    if (i < nb) {
      float2 t = aggs[i];
      a = seg_combine(a, Seg{t.x, t.y});
    }
  }
  Seg excl;
  (void)block_scan(a, s_wave, &excl);

  Seg run = excl;
  for (int g = 0; g < G; ++g) {
    int i = tid * G + g;
    if (i < nb) {
      pref[i] = make_float2(run.lo, run.hi);  // exclusive prefix per block
      float2 t = aggs[i];
      run = seg_combine(run, Seg{t.x, t.y});
    }
  }
}

// ---- pass 3: apply prefixes, emit outputs ---------------------------------
__global__ void __launch_bounds__(BLOCK)
k_apply(const float* __restrict__ x, const float* __restrict__ kw,
        const float* __restrict__ st, const float2* __restrict__ pref,
        float* __restrict__ out, long long T) {
  __shared__ float  s_x[CHUNK];
  __shared__ float2 s_wave[NWAVES];
  const int tid = threadIdx.x;
  const long long base = (long long)blockIdx.x * CHUNK;

  stage_chunk(s_x, x, base, T);   // x chunk served from the 192MB L2 here
  const float k  = kw[0];
  const float s0 = st[0];

  Seg a = seg_ident();
#pragma unroll
  for (int j = 0; j < ITEMS; ++j) {
    long long idx = base + (long long)tid * ITEMS + j;
    if (idx < T) {
      float y = k * s_x[tid * ITEMS + j];
      a = seg_combine(a, Seg{y, y + PLAY_WIDTH});
    }
  }
  Seg excl;
  (void)block_scan(a, s_wave, &excl);

  float2 pb = pref[blockIdx.x];
  Seg run = seg_combine(Seg{pb.x, pb.y}, excl);  // block prefix ∘ thread prefix
#pragma unroll
  for (int j = 0; j < ITEMS; ++j) {
    long long idx = base + (long long)tid * ITEMS + j;
    if (idx < T) {
      float y = k * s_x[tid * ITEMS + j];
      run = seg_combine(run, Seg{y, y + PLAY_WIDTH});
      float o = fminf(fmaxf(s0, run.lo), run.hi);
      __builtin_nontemporal_store(o, &out[idx]);   // NT: don't evict x from L2
    }
  }
}

// ---- host-side launcher ----------------------------------------------------
extern "C" void kernel_launch(void* const* d_in, const int* in_sizes, int n_in,
                              void* d_out, int out_size, void* d_ws, size_t ws_size,
                              hipStream_t stream) {
  (void)n_in; (void)out_size; (void)ws_size;
  const float* x  = (const float*)d_in[0];   // x: T floats
  const float* st = (const float*)d_in[1];   // state: scalar
  const float* kw = (const float*)d_in[2];   // kernel: scalar
  float* out = (float*)d_out;

  const long long T = (long long)in_sizes[0];
  const int nb = (int)((T + CHUNK - 1) / CHUNK);   // 4096 for T = 2^24

  float2* aggs = (float2*)d_ws;       // nb * 8 bytes
  float2* pref = aggs + nb;           // nb * 8 bytes

  k_block_agg<<<nb, BLOCK, 0, stream>>>(x, kw, aggs, T);
  k_scan_aggs<<<1,  BLOCK, 0, stream>>>(aggs, pref, nb);
  k_apply    <<<nb, BLOCK, 0, stream>>>(x, kw, st, pref, out, T);
}